// StateReprModuleP_5592047419688
// MI455X (gfx1250) — compile-verified
//
#include <hip/hip_runtime.h>

// ---------------------------------------------------------------------------
// StateRepr pairwise-interaction kernel for gfx1250 (MI455X).
// Store-bandwidth bound: 220 MB out / 23.3 TB/s ~ 10 us floor. No WMMA fit
// (elementwise Hadamard pair products, no K-contraction). Strategy:
//   gather -> LDS via CDNA5 GLOBAL_LOAD_ASYNC_TO_LDS_B128 (ASYNCcnt-tracked),
//   reuse each embedding ~19x from LDS, stream out with 128-bit NT stores.
// ---------------------------------------------------------------------------

#define N_ITEMS 20
#define EMB     64
#define NPAIR   190          // 20*19/2
#define NROW    210          // 20 singles + 190 pairs
#define THREADS 256

typedef float fv4 __attribute__((ext_vector_type(4)));
typedef int   v4i __attribute__((ext_vector_type(4)));

#if defined(__has_builtin)
#  if __has_builtin(__builtin_amdgcn_global_load_async_to_lds_b128) && \
      __has_builtin(__builtin_amdgcn_s_wait_asynccnt)
#    define USE_ASYNC_LDS 1
#  endif
#endif
#ifndef USE_ASYNC_LDS
#  define USE_ASYNC_LDS 0
#endif

// Builtin signature (from diagnostics): (v4i32 AS1*, v4i32 AS3*, imm, imm)
#define ASYNC_GPTR(p) ((__attribute__((address_space(1))) v4i*)(p))
#define ASYNC_LPTR(p) ((__attribute__((address_space(3))) v4i*)(p))

__global__ __launch_bounds__(THREADS) void
state_repr_kernel(const int* __restrict__ mem_idx,
                  const float* __restrict__ item_emb,
                  const float* __restrict__ weights,
                  float* __restrict__ out)
{
    __shared__ float          s_e[N_ITEMS][EMB];   // gathered embeddings (5 KB)
    __shared__ float          s_w[N_ITEMS];
    __shared__ int            s_idx[N_ITEMS];
    __shared__ float          s_wp[NPAIR];         // w[i1]*w[i2]
    __shared__ unsigned char  s_i1[NPAIR];
    __shared__ unsigned char  s_i2[NPAIR];

    const int b   = blockIdx.x;
    const int tid = threadIdx.x;
    const int d4  = tid & 15;   // float4 column: covers d = 4*d4 .. 4*d4+3
    const int rg  = tid >> 4;   // row group 0..15

    // ---- stage 0: indices + weights -> LDS -------------------------------
    if (tid < N_ITEMS) {
        s_idx[tid] = mem_idx[(size_t)b * N_ITEMS + tid];
        s_w[tid]   = weights[tid];
    }
    __syncthreads();

    // ---- stage 1: gather 20x64 fp32 tile into LDS ------------------------
    {
        const float* src0 = item_emb + (size_t)s_idx[rg] * EMB + d4 * 4;
#if USE_ASYNC_LDS
        __builtin_amdgcn_global_load_async_to_lds_b128(
            ASYNC_GPTR(src0), ASYNC_LPTR(&s_e[rg][d4 * 4]), 0, 0);
#else
        *(fv4*)&s_e[rg][d4 * 4] = *(const fv4*)src0;
#endif
        if (rg < N_ITEMS - 16) {  // rows 16..19
            const int r2 = 16 + rg;
            const float* src1 = item_emb + (size_t)s_idx[r2] * EMB + d4 * 4;
#if USE_ASYNC_LDS
            __builtin_amdgcn_global_load_async_to_lds_b128(
                ASYNC_GPTR(src1), ASYNC_LPTR(&s_e[r2][d4 * 4]), 0, 0);
#else
            *(fv4*)&s_e[r2][d4 * 4] = *(const fv4*)src1;
#endif
        }
    }

    // ---- stage 1b: build pair tables (overlaps with async gather) --------
    if (tid < NPAIR) {
        int p = tid, i1 = 0, rem = p;
        while (rem >= N_ITEMS - 1 - i1) { rem -= N_ITEMS - 1 - i1; ++i1; }
        const int i2 = i1 + 1 + rem;
        s_i1[p] = (unsigned char)i1;
        s_i2[p] = (unsigned char)i2;
        s_wp[p] = s_w[i1] * s_w[i2];
    }

#if USE_ASYNC_LDS
    __builtin_amdgcn_s_wait_asynccnt(0);   // my wave's async LDS writes done
#endif
    __syncthreads();                       // everyone's writes visible

    // ---- stage 2: emit 210 rows of 64 floats, 128b NT stores -------------
    float* outb = out + (size_t)b * (NROW * EMB);
    for (int r = rg; r < NROW; r += 16) {
        fv4 v;
        if (r < N_ITEMS) {
            v = *(const fv4*)&s_e[r][d4 * 4];
        } else {
            const int p  = r - N_ITEMS;
            const int i1 = s_i1[p];
            const int i2 = s_i2[p];
            const fv4 a  = *(const fv4*)&s_e[i1][d4 * 4];
            const fv4 c  = *(const fv4*)&s_e[i2][d4 * 4];
            v = (a * c) * s_wp[p];
        }
        // 220 MB output > 192 MB L2: use non-temporal stores.
        __builtin_nontemporal_store(v, (fv4*)(outb + r * EMB + d4 * 4));
    }
}

extern "C" void kernel_launch(void* const* d_in, const int* in_sizes, int n_in,
                              void* d_out, int out_size, void* d_ws, size_t ws_size,
                              hipStream_t stream) {
    // setup_inputs order: user(int32, unused), memory(int32 [B,N]),
    //                     item_emb(f32 [100001,64]), weights(f32 [N])
    const int*   mem_idx  = (const int*)d_in[1];
    const float* item_emb = (const float*)d_in[2];
    const float* weights  = (const float*)d_in[3];
    float*       out      = (float*)d_out;

    const int B = in_sizes[0];  // user count == batch
    state_repr_kernel<<<dim3(B), dim3(THREADS), 0, stream>>>(
        mem_idx, item_emb, weights, out);
}